// HybridSAGEClassifier_80470507258311
// MI455X (gfx1250) — compile-verified
//
#include <hip/hip_runtime.h>

// ---------------------------------------------------------------------------
// HybridSAGEClassifier for MI455X (gfx1250, wave32)
//   3x SAGEConv(mean) + BN + ReLU, then fusion MLP.
//   GEMMs: V_WMMA_F32_16X16X4_F32 (fp32 in/acc, matches fp32 reference),
//          A-tiles staged LDS via global_load_async_to_lds_b128 (ASYNCcnt),
//          weights staged LDS pre-swizzled into WMMA fragment order so every
//          fragment is one conflict-free ds_load_b64.
//   Edge aggregation: fp32 global atomics (traffic resolves in 192MB L2).
// ---------------------------------------------------------------------------

typedef float v2f __attribute__((ext_vector_type(2)));
typedef float v8f __attribute__((ext_vector_type(8)));

#define HID 64
#define INDIM 128
#define GEMM_WAVES 8            // 8 waves * 16 rows = 128 rows per block
#define APITCH (INDIM + 4)      // LDS row pitch for A tiles (bank-conflict pad)

// ---------------- utility kernels ----------------

__global__ void zero_kernel(float* __restrict__ p, long long n) {
    long long i = (long long)blockIdx.x * blockDim.x + threadIdx.x;
    long long stride = (long long)gridDim.x * blockDim.x;
    for (; i < n; i += stride) p[i] = 0.0f;
}

__global__ void degree_kernel(const long long* __restrict__ ei, float* __restrict__ cnt, int E) {
    int i = blockIdx.x * blockDim.x + threadIdx.x;
    if (i >= E) return;
    int dst = (int)ei[E + i];
    atomicAdd(&cnt[dst], 1.0f);
}

// scatter-add source-node features into agg[dst]; d is 128 or 64, 4 floats/thread
__global__ void scatter_kernel(const float* __restrict__ x, const long long* __restrict__ ei,
                               float* __restrict__ agg, int E, int d) {
    int d4 = d >> 2;
    long long tid = (long long)blockIdx.x * blockDim.x + threadIdx.x;
    if (tid >= (long long)E * d4) return;
    int e = (int)(tid / d4);
    int q = ((int)(tid % d4)) << 2;
    int src = (int)ei[e];
    int dst = (int)ei[E + e];
    const float4 v = *(const float4*)(x + (long long)src * d + q);
    float* a = agg + (long long)dst * d + q;
    atomicAdd(a + 0, v.x);
    atomicAdd(a + 1, v.y);
    atomicAdd(a + 2, v.z);
    atomicAdd(a + 3, v.w);
}

// ---------------- fused SAGE GEMM: out = (agg/cnt) @ Wl + b + xin @ Wr ------
// K = in_dim (128 or 64), output width fixed HID=64.
// Fragment layouts per ISA 7.12.2 (32-bit, wave32):
//   A 16x4: lane l<16 row l {K0,K1}; lanes 16-31 same rows {K2,K3}
//   B 4x16: lane n<16 col n {K0,K1}; lanes 16-31 same cols {K2,K3}
//   C/D:    VGPR j = row j (lanes 0-15) / row j+8 (lanes 16-31), col = lane&15
// LDS weight swizzle: pair (W[ko][c], W[ko+1][c]) for ko=4f+2h stored at
//   float index 256*f + 4*c + 2*h  -> one aligned ds_load_b64 per fragment,
//   banks {4lr,4lr+1} (h=0) and {4lr+2,4lr+3} (h=1): all 64 banks, no conflict.
__global__ __launch_bounds__(256) void sage_gemm_kernel(
    const float* __restrict__ agg, const float* __restrict__ cnt,
    const float* __restrict__ xin,
    const float* __restrict__ Wl, const float* __restrict__ bias,
    const float* __restrict__ Wr,
    float* __restrict__ out, int N, int K)
{
    __shared__ float lds_w[2 * INDIM * HID];            // 64 KB swizzled Wl|Wr
    __shared__ float lds_a[GEMM_WAVES * 16 * APITCH];   // per-wave A tiles

    const int tid  = threadIdx.x;
    const int wave = tid >> 5;
    const int lane = tid & 31;
    const int lr   = lane & 15;
    const int half = lane >> 4;
    const int rowBase = (blockIdx.x * GEMM_WAVES + wave) * 16;
    const bool active = (rowBase < N);   // wave-uniform (N % 16 == 0)

    float* aw = &lds_a[wave * 16 * APITCH];
    const unsigned awLds = (unsigned)(size_t)aw;   // LDS byte offset (low 32 bits)
    const int pitchB = (K + 4) * 4;

    // ---- issue async DMA of this wave's 16xK A-tile (pass 0: agg), overlaps
    //      with the weight staging below. Tile is contiguous in global memory.
    if (active) {
        const int row  = lr;          // lane -> row of the tile
        const int cpos = half;        // lane -> starting 16B chunk
        const unsigned gRow = (unsigned)((rowBase + row) * K) * 4u;
        for (int c = cpos; c < (K >> 2); c += 2) {
            unsigned ldst = awLds + (unsigned)(row * pitchB + c * 16);
            unsigned voff = gRow + (unsigned)(c * 16);
            asm volatile("global_load_async_to_lds_b128 %0, %1, %2"
                         :: "v"(ldst), "v"(voff), "s"(agg) : "memory");
        }
    }

    // ---- stage both weight matrices into LDS in fragment-swizzled order ----
    for (int i = tid; i < K * 32; i += 256) {
        int ko2 = i >> 6, c = i & 63;
        int dst = ((ko2 >> 1) << 8) + (c << 2) + ((ko2 & 1) << 1);
        float2 vl, vr;
        vl.x = Wl[(2 * ko2) * HID + c];     vl.y = Wl[(2 * ko2 + 1) * HID + c];
        vr.x = Wr[(2 * ko2) * HID + c];     vr.y = Wr[(2 * ko2 + 1) * HID + c];
        *(float2*)(lds_w + dst)           = vl;
        *(float2*)(lds_w + K * HID + dst) = vr;
    }
    __syncthreads();
    if (!active) return;

    const int r = rowBase + lr;
    const float invc = 1.0f / fmaxf(cnt[r], 1.0f);

    v8f acc0, acc1, acc2, acc3;
    {
        float b0 = bias[lr], b1 = bias[16 + lr], b2 = bias[32 + lr], b3 = bias[48 + lr];
        #pragma unroll
        for (int j = 0; j < 8; ++j) { acc0[j] = b0; acc1[j] = b1; acc2[j] = b2; acc3[j] = b3; }
    }

    const int aoff = lr * (K + 4) + 2 * half;   // A fragment float offset
    const int boff = 4 * lr + 2 * half;         // B fragment float offset in swizzle

    #pragma unroll
    for (int pass = 0; pass < 2; ++pass) {
        if (pass == 1) {
            // A-tile reuse: drain our ds reads, then DMA the root-feature tile.
            asm volatile("s_wait_dscnt 0" ::: "memory");
            const int row  = lr;
            const int cpos = half;
            const unsigned gRow = (unsigned)((rowBase + row) * K) * 4u;
            for (int c = cpos; c < (K >> 2); c += 2) {
                unsigned ldst = awLds + (unsigned)(row * pitchB + c * 16);
                unsigned voff = gRow + (unsigned)(c * 16);
                asm volatile("global_load_async_to_lds_b128 %0, %1, %2"
                             :: "v"(ldst), "v"(voff), "s"(xin) : "memory");
            }
        }
        asm volatile("s_wait_asynccnt 0" ::: "memory");

        const float* wb = lds_w + (pass ? K * HID : 0);
        const float scale = (pass == 0 ? invc : 1.0f);
        for (int k = 0; k < K; k += 4) {
            v2f a = *(const v2f*)(aw + aoff + k);
            a = a * scale;
            const float* bbase = wb + 64 * k + boff;   // 256*(k/4) + boff
            v2f b0 = *(const v2f*)(bbase);
            v2f b1 = *(const v2f*)(bbase + 64);
            v2f b2 = *(const v2f*)(bbase + 128);
            v2f b3 = *(const v2f*)(bbase + 192);
            acc0 = __builtin_amdgcn_wmma_f32_16x16x4_f32(false, a, false, b0, (short)0, acc0, false, false);
            acc1 = __builtin_amdgcn_wmma_f32_16x16x4_f32(false, a, false, b1, (short)0, acc1, false, false);
            acc2 = __builtin_amdgcn_wmma_f32_16x16x4_f32(false, a, false, b2, (short)0, acc2, false, false);
            acc3 = __builtin_amdgcn_wmma_f32_16x16x4_f32(false, a, false, b3, (short)0, acc3, false, false);
        }
    }

    float* O = out + (long long)rowBase * HID;
    #pragma unroll
    for (int j = 0; j < 8; ++j) {
        int row = j + 8 * half;
        O[row * HID + lr]      = acc0[j];
        O[row * HID + 16 + lr] = acc1[j];
        O[row * HID + 32 + lr] = acc2[j];
        O[row * HID + 48 + lr] = acc3[j];
    }
}

// ---------------- batch-norm ----------------

__global__ void bn_stats_kernel(const float* __restrict__ h, float* __restrict__ stats, int N) {
    // stats[0:64]=sum, stats[64:128]=sum of squares (pre-zeroed)
    int c = threadIdx.x & 63;
    int rpb = blockDim.x >> 6;
    int row = blockIdx.x * rpb + (threadIdx.x >> 6);
    int rstride = gridDim.x * rpb;
    float s = 0.0f, s2 = 0.0f;
    for (int r = row; r < N; r += rstride) {
        float v = h[(long long)r * HID + c];
        s += v; s2 += v * v;
    }
    atomicAdd(&stats[c], s);
    atomicAdd(&stats[64 + c], s2);
}

__global__ void bn_apply_relu_kernel(float* __restrict__ h, const float* __restrict__ stats,
                                     const float* __restrict__ g, const float* __restrict__ be, int N) {
    long long i = (long long)blockIdx.x * blockDim.x + threadIdx.x;
    long long total = (long long)N * HID;
    if (i >= total) return;
    int c = (int)(i & 63);
    float inv_n = 1.0f / (float)N;
    float mu  = stats[c] * inv_n;
    float var = stats[64 + c] * inv_n - mu * mu;
    float v = (h[i] - mu) * rsqrtf(var + 1e-5f) * g[c] + be[c];
    h[i] = fmaxf(v, 0.0f);
}

// ---------------- fusion MLP: Linear(65,64)+ReLU -> Linear(64,1) ------------

__global__ void fusion_kernel(const float* __restrict__ h, const float* __restrict__ xgb,
                              const float* __restrict__ Wf1, const float* __restrict__ bf1,
                              const float* __restrict__ Wf2, const float* __restrict__ bf2,
                              float* __restrict__ out, int N) {
    int i = blockIdx.x * blockDim.x + threadIdx.x;
    if (i >= N) return;
    float xg = xgb[i];
    float z[64];
    #pragma unroll
    for (int j = 0; j < 64; ++j) z[j] = bf1[j] + xg * Wf1[64 * 64 + j];
    const float* hr = h + (long long)i * HID;
    for (int k = 0; k < 64; ++k) {
        float hk = hr[k];
        const float* wrow = Wf1 + k * 64;
        #pragma unroll
        for (int j = 0; j < 64; ++j) z[j] += hk * wrow[j];
    }
    float acc = bf2[0];
    #pragma unroll
    for (int j = 0; j < 64; ++j) acc += fmaxf(z[j], 0.0f) * Wf2[j];
    out[i] = acc;
}

// ---------------- host-side launch sequence ----------------

extern "C" void kernel_launch(void* const* d_in, const int* in_sizes, int n_in,
                              void* d_out, int out_size, void* d_ws, size_t ws_size,
                              hipStream_t stream) {
    const float*     x    = (const float*)d_in[0];
    const long long* ei   = (const long long*)d_in[1];
    const float*     xgb  = (const float*)d_in[2];
    const float* W1l = (const float*)d_in[3];
    const float* b1  = (const float*)d_in[4];
    const float* W1r = (const float*)d_in[5];
    const float* g1  = (const float*)d_in[6];
    const float* be1 = (const float*)d_in[7];
    const float* W2l = (const float*)d_in[8];
    const float* b2  = (const float*)d_in[9];
    const float* W2r = (const float*)d_in[10];
    const float* g2  = (const float*)d_in[11];
    const float* be2 = (const float*)d_in[12];
    const float* W3l = (const float*)d_in[13];
    const float* b3  = (const float*)d_in[14];
    const float* W3r = (const float*)d_in[15];
    const float* g3  = (const float*)d_in[16];
    const float* be3 = (const float*)d_in[17];
    const float* Wf1 = (const float*)d_in[18];
    const float* bf1 = (const float*)d_in[19];
    const float* Wf2 = (const float*)d_in[20];
    const float* bf2 = (const float*)d_in[21];
    float* out = (float*)d_out;

    const int N = in_sizes[0] / INDIM;
    const int E = in_sizes[1] / 2;

    float* ws    = (float*)d_ws;
    float* cnt   = ws;
    float* agg   = cnt + N;                       // N*128 floats
    float* hA    = agg + (long long)N * INDIM;    // N*64
    float* hB    = hA  + (long long)N * HID;      // N*64
    float* stats = hB  + (long long)N * HID;      // 128

    const int T = 256;
    const int gemmBlocks = (N + 16 * GEMM_WAVES - 1) / (16 * GEMM_WAVES);
    const int bnApplyBlocks = (int)(((long long)N * HID + T - 1) / T);
    const int statsBlocks = 1024;

    // ---- layer 1 (in=128) ----
    zero_kernel<<<2048, T, 0, stream>>>(cnt, N);
    zero_kernel<<<4096, T, 0, stream>>>(agg, (long long)N * INDIM);
    degree_kernel<<<(E + T - 1) / T, T, 0, stream>>>(ei, cnt, E);
    {
        long long work = (long long)E * (INDIM / 4);
        scatter_kernel<<<(int)((work + T - 1) / T), T, 0, stream>>>(x, ei, agg, E, INDIM);
    }
    sage_gemm_kernel<<<gemmBlocks, T, 0, stream>>>(agg, cnt, x, W1l, b1, W1r, hA, N, INDIM);
    zero_kernel<<<1, 128, 0, stream>>>(stats, 128);
    bn_stats_kernel<<<statsBlocks, T, 0, stream>>>(hA, stats, N);
    bn_apply_relu_kernel<<<bnApplyBlocks, T, 0, stream>>>(hA, stats, g1, be1, N);

    // ---- layer 2 (in=64) ----
    zero_kernel<<<4096, T, 0, stream>>>(agg, (long long)N * HID);
    {
        long long work = (long long)E * (HID / 4);
        scatter_kernel<<<(int)((work + T - 1) / T), T, 0, stream>>>(hA, ei, agg, E, HID);
    }
    sage_gemm_kernel<<<gemmBlocks, T, 0, stream>>>(agg, cnt, hA, W2l, b2, W2r, hB, N, HID);
    zero_kernel<<<1, 128, 0, stream>>>(stats, 128);
    bn_stats_kernel<<<statsBlocks, T, 0, stream>>>(hB, stats, N);
    bn_apply_relu_kernel<<<bnApplyBlocks, T, 0, stream>>>(hB, stats, g2, be2, N);

    // ---- layer 3 (in=64) ----
    zero_kernel<<<4096, T, 0, stream>>>(agg, (long long)N * HID);
    {
        long long work = (long long)E * (HID / 4);
        scatter_kernel<<<(int)((work + T - 1) / T), T, 0, stream>>>(hB, ei, agg, E, HID);
    }
    sage_gemm_kernel<<<gemmBlocks, T, 0, stream>>>(agg, cnt, hB, W3l, b3, W3r, hA, N, HID);
    zero_kernel<<<1, 128, 0, stream>>>(stats, 128);
    bn_stats_kernel<<<statsBlocks, T, 0, stream>>>(hA, stats, N);
    bn_apply_relu_kernel<<<bnApplyBlocks, T, 0, stream>>>(hA, stats, g3, be3, N);

    // ---- fusion MLP ----
    fusion_kernel<<<(N + T - 1) / T, T, 0, stream>>>(hA, xgb, Wf1, bf1, Wf2, bf2, out, N);
}